// YoloLossLayer_43731357007998
// MI455X (gfx1250) — compile-verified
//
#include <hip/hip_runtime.h>
#include <cstdint>
#include <cstddef>

// ---------------- constants from the reference ----------------
#define CLS          80
#define STRIDE       85          // 5 + CLASS_NUM
#define IGN_THR      0.5f
#define EPSF         1e-16f
#define CONF_BLOCKS  1024
#define RED_T        256

#ifndef __has_builtin
#define __has_builtin(x) 0
#endif
#if __has_builtin(__builtin_amdgcn_global_load_async_to_lds_b128)
#define USE_ASYNC_LDS 1
#else
#define USE_ASYNC_LDS 0
#endif

typedef __attribute__((ext_vector_type(2))) float v2f;
typedef __attribute__((ext_vector_type(8))) float v8f;
typedef __attribute__((ext_vector_type(4))) int   v4i;

__device__ __forceinline__ float softplusf(float x) {
  // numerically stable softplus
  return fmaxf(x, 0.f) + log1pf(__expf(-fabsf(x)));
}

__device__ __forceinline__ float block_reduce_sum(float v) {
  __shared__ float red[RED_T];
  int tid = threadIdx.x;
  red[tid] = v;
  __syncthreads();
  for (int s = RED_T >> 1; s > 0; s >>= 1) {
    if (tid < s) red[tid] += red[tid + s];
    __syncthreads();
  }
  return red[0];
}

// -------- kernel 1: per-target anchor match, scatter obj bitmask, dedup -----
__global__ __launch_bounds__(RED_T)
void k_prep(const float* __restrict__ priori, const int* __restrict__ fm,
            const float* __restrict__ targets, int B, int N, int T,
            int* __restrict__ tgt_lin, int* __restrict__ tgt_lab,
            int* __restrict__ tgt_owner, float* __restrict__ tbox,
            float* __restrict__ txyxy, unsigned int* __restrict__ objmask,
            int maskWords) {
  int tid = threadIdx.x;
  for (int i = tid; i < maskWords; i += blockDim.x) objmask[i] = 0u;
  __syncthreads();
  if (tid < T) {
    int fmv[3] = {fm[0], fm[1], fm[2]};
    int offs[9]; int acc = 0;
    for (int i = 0; i < 9; ++i) { offs[i] = acc; int f = fmv[i / 3]; acc += f * f; }
    const float* tg = targets + (size_t)tid * 6;
    int   tb  = (int)tg[0];
    int   lab = (int)tg[1];
    float cx = tg[2], cy = tg[3], tw = tg[4], th = tg[5];
    float best = -1.f; int ind = 0;
    for (int a = 0; a < 9; ++a) {
      float aw = priori[(size_t)offs[a] * 4 + 2];
      float ah = priori[(size_t)offs[a] * 4 + 3];
      float inter = fminf(tw, aw) * fminf(th, ah);
      float iou = inter / (tw * th + aw * ah - inter + EPSF);
      if (iou > best) { best = iou; ind = a; }     // first-max like jnp.argmax
    }
    int m  = fmv[ind / 3];
    int hh = (int)(cy * (float)m);
    int ww = (int)(cx * (float)m);
    int pidx = offs[ind] + ind * m * m + hh * m + ww;
    int lin  = tb * N + pidx;
    tgt_lin[tid] = lin;
    tgt_lab[tid] = lab;
    tbox[tid * 4 + 0] = cx; tbox[tid * 4 + 1] = cy;
    tbox[tid * 4 + 2] = tw; tbox[tid * 4 + 3] = th;
    txyxy[tid * 4 + 0] = cx - tw * 0.5f;
    txyxy[tid * 4 + 1] = cy - th * 0.5f;
    txyxy[tid * 4 + 2] = cx + tw * 0.5f;
    txyxy[tid * 4 + 3] = cy + th * 0.5f;
    atomicOr(&objmask[lin >> 5], 1u << (lin & 31));
  }
  __syncthreads();
  if (tid < T) {  // deterministic dedup: lowest target index owns a position
    int mylin = tgt_lin[tid];
    int owner = 1;
    for (int t2 = 0; t2 < tid; ++t2)
      if (tgt_lin[t2] == mylin) { owner = 0; break; }
    tgt_owner[tid] = owner;
  }
}

// -------- kernel 2: ignore[n] = any_t IoU(target_t, prior_n) > 0.5 ----------
__global__ __launch_bounds__(RED_T)
void k_ignore(const float* __restrict__ priori, const float* __restrict__ txyxy,
              int N, int T, unsigned char* __restrict__ ignore_) {
  __shared__ float sxy[256 * 4];
  __shared__ float sarea[256];
  int tid = threadIdx.x;
#if USE_ASYNC_LDS
  if (tid < T) {
    // async copy of the 16B target box straight into LDS (CDNA5 async path)
    __builtin_amdgcn_global_load_async_to_lds_b128(
        (__attribute__((address_space(1))) v4i*)(txyxy + (size_t)tid * 4),
        (__attribute__((address_space(3))) v4i*)&sxy[tid * 4], 0, 0);
  }
#if __has_builtin(__builtin_amdgcn_s_wait_asynccnt)
  __builtin_amdgcn_s_wait_asynccnt(0);
#else
  asm volatile("s_wait_asynccnt 0" ::: "memory");
#endif
#else
  for (int i = tid; i < T * 4; i += blockDim.x) sxy[i] = txyxy[i];
#endif
  __syncthreads();
  if (tid < T)
    sarea[tid] = (sxy[tid * 4 + 2] - sxy[tid * 4 + 0]) *
                 (sxy[tid * 4 + 3] - sxy[tid * 4 + 1]);
  __syncthreads();
  int n = blockIdx.x * blockDim.x + tid;
  if (n >= N) return;
  float4 p = ((const float4*)priori)[n];
  float bx0 = p.x - p.z * 0.5f, by0 = p.y - p.w * 0.5f;
  float bx1 = p.x + p.z * 0.5f, by1 = p.y + p.w * 0.5f;
  float areaB = (bx1 - bx0) * (by1 - by0);
  unsigned char ig = 0;
  for (int t = 0; t < T; ++t) {
    float iw = fmaxf(fminf(sxy[t * 4 + 2], bx1) - fmaxf(sxy[t * 4 + 0], bx0), 0.f);
    float ih = fmaxf(fminf(sxy[t * 4 + 3], by1) - fmaxf(sxy[t * 4 + 1], by0), 0.f);
    float inter = iw * ih;
    float iou = inter / (sarea[t] + areaB - inter + EPSF);
    if (iou > IGN_THR) { ig = 1; break; }
  }
  ignore_[n] = ig;
}

// -------- kernel 3: confidence loss over all (b,n) — the bandwidth kernel ---
__global__ __launch_bounds__(RED_T)
void k_conf(const float* __restrict__ preds, const unsigned int* __restrict__ objmask,
            const unsigned char* __restrict__ ignore_, int BN, int N,
            float* __restrict__ partials) {
  float acc = 0.f;
  for (int lin = blockIdx.x * blockDim.x + threadIdx.x; lin < BN;
       lin += gridDim.x * blockDim.x) {
    float x = preds[(size_t)lin * STRIDE + 4];   // only the conf plane is read
    int  n  = lin % N;
    bool obj = (objmask[lin >> 5] >> (lin & 31)) & 1u;
    float s  = 1.f / (1.f + __expf(-x));
    float sp = softplusf(x);
    if (obj) {
      float om = 1.f - s;
      acc += 0.25f * om * om * (sp - x);         // focal pos, target=1
    } else if (!ignore_[n]) {
      acc += s * s * sp;                         // focal neg, target=0
    }
  }
  float tot = block_reduce_sum(acc);
  if (threadIdx.x == 0) partials[blockIdx.x] = tot;
}

// -------- kernel 4: class BCE at unique obj positions -----------------------
__global__ __launch_bounds__(RED_T)
void k_cls(const float* __restrict__ preds, const int* __restrict__ tgt_lin,
           const int* __restrict__ tgt_lab, const int* __restrict__ tgt_owner,
           int T, float* __restrict__ partials, int slot) {
  float acc = 0.f;
  int total = T * CLS;
  for (int i = threadIdx.x; i < total; i += blockDim.x) {
    int t = i / CLS, c = i % CLS;
    if (!tgt_owner[t]) continue;
    float x = preds[(size_t)tgt_lin[t] * STRIDE + 5 + c];
    acc += softplusf(x) - ((c == tgt_lab[t]) ? x : 0.f);
  }
  float tot = block_reduce_sum(acc);
  if (threadIdx.x == 0) partials[slot] = tot;
}

// -------- kernel 5: paired GIoU loss over all T targets ---------------------
__global__ __launch_bounds__(RED_T)
void k_giou(const float* __restrict__ preds, const int* __restrict__ tgt_lin,
            const float* __restrict__ tbox, int T,
            float* __restrict__ partials, int slot) {
  float acc = 0.f;
  int t = threadIdx.x;
  if (t < T) {
    const float* pb = preds + (size_t)tgt_lin[t] * STRIDE;
    float p0 = pb[0], p1 = pb[1], p2 = pb[2], p3 = pb[3];
    float q0 = tbox[t * 4 + 0], q1 = tbox[t * 4 + 1];
    float q2 = tbox[t * 4 + 2], q3 = tbox[t * 4 + 3];
    float ax0 = p0 - p2 * 0.5f, ay0 = p1 - p3 * 0.5f;
    float ax1 = p0 + p2 * 0.5f, ay1 = p1 + p3 * 0.5f;
    float bx0 = q0 - q2 * 0.5f, by0 = q1 - q3 * 0.5f;
    float bx1 = q0 + q2 * 0.5f, by1 = q1 + q3 * 0.5f;
    float iw = fmaxf(fminf(ax1, bx1) - fmaxf(ax0, bx0), 0.f);
    float ih = fmaxf(fminf(ay1, by1) - fmaxf(ay0, by0), 0.f);
    float inter = iw * ih;
    float uni = p2 * p3 + q2 * q3 - inter;
    float iou = inter / (uni + EPSF);
    float cw = fmaxf(fmaxf(ax1, bx1) - fminf(ax0, bx0), 0.f);
    float ch = fmaxf(fmaxf(ay1, by1) - fminf(ay0, by0), 0.f);
    float carea = cw * ch;
    float giou = iou - (carea - uni) / (carea + EPSF);
    acc = (2.f - q2 * q3) * (1.f - giou);
  }
  float tot = block_reduce_sum(acc);
  if (threadIdx.x == 0) partials[slot] = tot;
}

// -------- kernel 6: single-wave WMMA reduction of all partial sums ----------
// A holds 64 partials (16x4 f32), B is all-ones (4x16): D[m][n] = rowsum_m + C.
// Chained over chunks; grand total = sum_m D[m][0] = (lane0 + lane16) per-lane
// sums of the 8 accumulator VGPRs (C/D layout: VGPRi lane l<16 -> M=i,N=l;
// lane>=16 -> M=8+i,N=l-16).
__global__ __launch_bounds__(32)
void k_final(const float* __restrict__ partials, int ptotal, float invB,
             float* __restrict__ out) {
  int lane = threadIdx.x;
  v8f c = {0.f, 0.f, 0.f, 0.f, 0.f, 0.f, 0.f, 0.f};
  v2f ones; ones.x = 1.f; ones.y = 1.f;
  int nch = (ptotal + 63) / 64;
  for (int ch = 0; ch < nch; ++ch) {
    int i0 = ch * 64 + lane;
    int i1 = ch * 64 + 32 + lane;
    v2f a;
    a.x = (i0 < ptotal) ? partials[i0] : 0.f;
    a.y = (i1 < ptotal) ? partials[i1] : 0.f;
    c = __builtin_amdgcn_wmma_f32_16x16x4_f32(false, a, false, ones,
                                              (short)0, c, false, false);
  }
  float v = c[0] + c[1] + c[2] + c[3] + c[4] + c[5] + c[6] + c[7];
  float tot = __shfl(v, 0, 32) + __shfl(v, 16, 32);
  if (lane == 0) out[0] = tot * invB;
}

// ---------------------------------------------------------------------------
extern "C" void kernel_launch(void* const* d_in, const int* in_sizes, int n_in,
                              void* d_out, int out_size, void* d_ws, size_t ws_size,
                              hipStream_t stream) {
  const float* preds   = (const float*)d_in[0];
  const float* priori  = (const float*)d_in[1];
  const int*   fm      = (const int*)d_in[2];
  const float* targets = (const float*)d_in[4];   // d_in[3] = input_dim, unused

  int N  = in_sizes[1] / 4;
  int B  = in_sizes[0] / (N * STRIDE);
  int T  = in_sizes[4] / 6;
  int BN = B * N;
  int maskWords = (BN + 31) / 32;

  char* w = (char*)d_ws;
  size_t off = 0;
  auto take = [&](size_t bytes) {
    char* p = w + off;
    off = (off + bytes + 15) & ~(size_t)15;
    return p;
  };
  int*          tgt_lin   = (int*)take((size_t)T * 4);
  int*          tgt_lab   = (int*)take((size_t)T * 4);
  int*          tgt_owner = (int*)take((size_t)T * 4);
  float*        tbox      = (float*)take((size_t)T * 16);
  float*        txyxy     = (float*)take((size_t)T * 16);
  unsigned int* objmask   = (unsigned int*)take((size_t)maskWords * 4);
  unsigned char* ignore_  = (unsigned char*)take((size_t)N);
  int ptotal = CONF_BLOCKS + 2;
  float*        partials  = (float*)take((size_t)ptotal * 4);
  (void)ws_size; (void)n_in; (void)out_size;

  k_prep<<<1, RED_T, 0, stream>>>(priori, fm, targets, B, N, T, tgt_lin,
                                  tgt_lab, tgt_owner, tbox, txyxy, objmask,
                                  maskWords);
  k_ignore<<<(N + RED_T - 1) / RED_T, RED_T, 0, stream>>>(priori, txyxy, N, T,
                                                          ignore_);
  k_conf<<<CONF_BLOCKS, RED_T, 0, stream>>>(preds, objmask, ignore_, BN, N,
                                            partials);
  k_cls<<<1, RED_T, 0, stream>>>(preds, tgt_lin, tgt_lab, tgt_owner, T,
                                 partials, CONF_BLOCKS);
  k_giou<<<1, RED_T, 0, stream>>>(preds, tgt_lin, tbox, T, partials,
                                  CONF_BLOCKS + 1);
  k_final<<<1, 32, 0, stream>>>(partials, ptotal, 1.f / (float)B,
                                (float*)d_out);
}